// PEPS_9740985827486
// MI455X (gfx1250) — compile-verified
//
#include <hip/hip_runtime.h>
#include <math.h>

// ---------------- problem constants ----------------
#define L1C 6
#define L2C 6
#define DBo 6          // bond dim
#define DPH 2          // phys dim
#define DCC 24         // truncation chi
#define BATCHC 32
#define NTHR 256
#define NWAVE 8

typedef float v2f __attribute__((ext_vector_type(2)));
typedef float v8f __attribute__((ext_vector_type(8)));
typedef unsigned int v4u __attribute__((ext_vector_type(4)));
typedef int v4i __attribute__((ext_vector_type(4)));
typedef int v8i __attribute__((ext_vector_type(8)));

#if defined(__has_builtin)
#  if __has_builtin(__builtin_amdgcn_tensor_load_to_lds) && __has_builtin(__builtin_amdgcn_s_wait_tensorcnt)
#    define HAVE_TDM 1
#  else
#    define HAVE_TDM 0
#  endif
#else
#  define HAVE_TDM 0
#endif

// ---------------- per-batch scratch layout (floats) ----------------
constexpr int SITE_SZ  = DBo * DBo * DBo * DBo;  // 1296
constexpr int DBL_CAP  = 124416;                 // 144*6*24*6 max double tensor
constexpr int RG_CAP   = 20736;                  // 144*144 (R / Gram / U / Vh temp)
constexpr int W_CAP    = 4096;                   // 144*24 padded
constexpr int CAND_CAP = 4096;                   // 24*6*24 padded

constexpr int OFF_DBL  = 0;
constexpr int OFF_TMP  = OFF_DBL + 6 * DBL_CAP;
constexpr int OFF_R    = OFF_TMP + DBL_CAP;
constexpr int OFF_G    = OFF_R   + RG_CAP;
constexpr int OFF_U    = OFF_G   + RG_CAP;
constexpr int OFF_W    = OFF_U   + RG_CAP;
constexpr int OFF_CAND = OFF_W   + W_CAP;
constexpr int PER_BATCH = OFF_CAND + 6 * CAND_CAP;   // ~962k floats; *32 batches ~123MB -> L2 resident

// ---------------- TDM: async 1-D global->LDS copy (CDNA5 Tensor Data Mover) ---
// Builds a D# per CDNA5 ISA ch.8: group0 {count=1, lds_addr, global_addr, type=2},
// group1 {data_size=4B, tensor_dim0 = tile_dim0 = nelem, stride0 = nelem}.
// Issued by one wave (TDM ignores EXEC); completion via TENSORcnt.
__device__ inline void tdm_load_lds_1d(void* lds_dst, const void* gsrc, unsigned nelem)
{
#if HAVE_TDM
    const unsigned long long ga = (unsigned long long)(uintptr_t)gsrc;
    const unsigned lds = (unsigned)(uintptr_t)lds_dst;   // LDS aperture: addr[31:0] = LDS offset
    v4u g0;
    g0[0] = 1u;                                              // count=1 (valid user D#)
    g0[1] = lds;                                             // lds_addr
    g0[2] = (unsigned)(ga & 0xFFFFFFFFu);                    // global_addr[31:0]
    g0[3] = (unsigned)((ga >> 32) & 0x01FFFFFFu) | (2u << 30); // global_addr[56:32] | type=2
    v8i g1;
    g1[0] = (int)(2u << 16);                                 // wg_mask=0, data_size=2 (4 bytes)
    g1[1] = (int)((nelem & 0xFFFFu) << 16);                  // tensor_dim0[15:0]
    g1[2] = (int)(((nelem >> 16) & 0xFFFFu) | (1u << 16));   // tensor_dim0[31:16], tensor_dim1=1
    g1[3] = (int)((nelem & 0xFFFFu) << 16);                  // tile_dim0 = nelem
    g1[4] = 0;                                               // tile_dim1=0 (1-D), tile_dim2=0
    g1[5] = (int)nelem;                                      // tensor_dim0_stride[31:0]
    g1[6] = 0;
    g1[7] = 0;
    v4i gz = {0, 0, 0, 0};
#  if __clang_major__ >= 23
    v8i gz8 = {0, 0, 0, 0, 0, 0, 0, 0};
    __builtin_amdgcn_tensor_load_to_lds(g0, g1, gz, gz, gz8, 0);
#  else
    __builtin_amdgcn_tensor_load_to_lds(g0, g1, gz, gz, 0);
#  endif
    __builtin_amdgcn_s_wait_tensorcnt(0);
#else
    (void)lds_dst; (void)gsrc; (void)nelem;
#endif
}

// ---------------- block reduction (wave shuffle + LDS) ----------------
__device__ inline float block_sum(float v, volatile float* red, int tid)
{
    for (int off = 16; off > 0; off >>= 1) v += __shfl_down(v, off, 32);
    if ((tid & 31) == 0) red[tid >> 5] = v;
    __syncthreads();
    if (tid == 0) {
        float s = 0.f;
        for (int i = 0; i < NWAVE; ++i) s += red[i];
        red[0] = s;
    }
    __syncthreads();
    float s = red[0];
    __syncthreads();
    return s;
}

// ---------------- generic dynamic-shape GEMM on the f32 WMMA pipe ----------
// C(MxN) = opA(A)*opB(B), row-major, 16x16 tiles over 8 waves, K in steps of 4
// via V_WMMA_F32_16X16X4_F32. Interior tiles: unconditional loads/stores (no
// EXEC writes at all). Edge tiles/K-tail: clamp-in-bounds loads + v_cndmask
// zero-selects — still branchless, EXEC stays all-ones as WMMA requires.
__device__ inline void gemm_wmma(const float* __restrict__ A, int lda, int tA,
                                 const float* __restrict__ B, int ldb, int tB,
                                 float* __restrict__ C, int ldc,
                                 int M, int N, int K, int tid)
{
    const int lane = tid & 31;
    const int wv   = tid >> 5;
    const int half = lane >> 4;          // 0: lanes 0-15, 1: lanes 16-31
    const int lrow = lane & 15;
    const int tilesN = (N + 15) >> 4;
    const int tiles  = ((M + 15) >> 4) * tilesN;
    for (int t = wv; t < tiles; t += NWAVE) {
        const int tm = (t / tilesN) << 4;
        const int tn = (t % tilesN) << 4;
        const bool fullM = (tm + 16 <= M);
        const bool fullN = (tn + 16 <= N);
        const int ar  = tm + lrow;
        const int bc  = tn + lrow;
        const int arc = (ar < M) ? ar : (M - 1);
        const int bcc = (bc < N) ? bc : (N - 1);
        v8f acc = {};
        int kk = 0;
        if (fullM && fullN) {            // wave-uniform branch: s_cbranch only
            for (; kk + 4 <= K; kk += 4) {
                const int ka = kk + half * 2;
                v2f af, bf;
                af.x = tA ? A[(size_t)ka * lda + ar]       : A[(size_t)ar * lda + ka];
                af.y = tA ? A[(size_t)(ka + 1) * lda + ar] : A[(size_t)ar * lda + ka + 1];
                bf.x = tB ? B[(size_t)bc * ldb + ka]       : B[(size_t)ka * ldb + bc];
                bf.y = tB ? B[(size_t)bc * ldb + ka + 1]   : B[(size_t)(ka + 1) * ldb + bc];
                acc = __builtin_amdgcn_wmma_f32_16x16x4_f32(
                          false, af, false, bf, (short)0, acc, false, false);
            }
        }
        for (; kk < K; kk += 4) {        // edge tiles and K tail: clamped, branchless
            const int ka  = kk + half * 2;
            const int ka0 = (ka     < K) ? ka     : (K - 1);
            const int ka1 = (ka + 1 < K) ? ka + 1 : (K - 1);
            const float a0 = tA ? A[(size_t)ka0 * lda + arc] : A[(size_t)arc * lda + ka0];
            const float a1 = tA ? A[(size_t)ka1 * lda + arc] : A[(size_t)arc * lda + ka1];
            const float b0 = tB ? B[(size_t)bcc * ldb + ka0] : B[(size_t)ka0 * ldb + bcc];
            const float b1 = tB ? B[(size_t)bcc * ldb + ka1] : B[(size_t)ka1 * ldb + bcc];
            v2f af, bf;
            af.x = (ar < M && ka     < K) ? a0 : 0.f;
            af.y = (ar < M && ka + 1 < K) ? a1 : 0.f;
            bf.x = (bc < N && ka     < K) ? b0 : 0.f;
            bf.y = (bc < N && ka + 1 < K) ? b1 : 0.f;
            acc = __builtin_amdgcn_wmma_f32_16x16x4_f32(
                      false, af, false, bf, (short)0, acc, false, false);
        }
        if (fullM && fullN) {
#pragma unroll
            for (int v = 0; v < 8; ++v)
                C[(size_t)(tm + v + half * 8) * ldc + tn + lrow] = acc[v];
        } else {
#pragma unroll
            for (int v = 0; v < 8; ++v) {
                const int r = tm + v + half * 8;
                const int c = tn + lrow;
                if (r < M && c < N) C[(size_t)r * ldc + c] = acc[v];
            }
        }
    }
}

// ---------------- in-place modified Gram-Schmidt QR -----------------------
__device__ inline void mgs_qr(float* A, int m, int n, int k, float* Rb,
                              volatile float* red, int tid)
{
    for (int i = tid; i < k * n; i += NTHR) Rb[i] = 0.f;
    __syncthreads();
    const int lane = tid & 31, wv = tid >> 5;
    for (int j = 0; j < k; ++j) {
        float s = 0.f;
        for (int i = tid; i < m; i += NTHR) { float v = A[i * n + j]; s += v * v; }
        s = block_sum(s, red, tid);
        const float nrm = sqrtf(s);
        const float inv = (nrm > 1e-30f) ? 1.f / nrm : 0.f;
        if (tid == 0) Rb[j * n + j] = nrm;
        for (int i = tid; i < m; i += NTHR) A[i * n + j] *= inv;
        __syncthreads();
        for (int c = j + 1 + wv; c < n; c += NWAVE) {   // one wave per column
            float t = 0.f;
            for (int i = lane; i < m; i += 32) t += A[i * n + j] * A[i * n + c];
            for (int off = 16; off > 0; off >>= 1) t += __shfl_down(t, off, 32);
            t = __shfl(t, 0, 32);
            for (int i = lane; i < m; i += 32) A[i * n + c] -= t * A[i * n + j];
            if (lane == 0) Rb[j * n + c] = t;
        }
        __syncthreads();
    }
}

// ---------------- two-sided Jacobi eigensolver on symmetric G (l x l) -----
__device__ inline void jacobi_eig(float* G, float* U, int l,
                                  volatile float* sbc, int tid)
{
    for (int i = tid; i < l * l; i += NTHR) U[i] = ((i / l) == (i % l)) ? 1.f : 0.f;
    __syncthreads();
    for (int sweep = 0; sweep < 5; ++sweep) {
        for (int p = 0; p < l - 1; ++p) {
            for (int q = p + 1; q < l; ++q) {
                if (tid == 0) {
                    const float app = G[p * l + p], aqq = G[q * l + q], apq = G[p * l + q];
                    float c = 1.f, s = 0.f;
                    if (fabsf(apq) > 1e-24f) {
                        const float tau = (aqq - app) / (2.f * apq);
                        const float tt  = ((tau >= 0.f) ? 1.f : -1.f) /
                                          (fabsf(tau) + sqrtf(1.f + tau * tau));
                        c = rsqrtf(1.f + tt * tt);
                        s = tt * c;
                    }
                    sbc[0] = c; sbc[1] = s;
                }
                __syncthreads();
                const float c = sbc[0], s = sbc[1];
                if (s != 0.f) {                    // uniform branch (LDS broadcast)
                    for (int i = tid; i < l; i += NTHR) {
                        const float gp = G[i * l + p], gq = G[i * l + q];
                        G[i * l + p] = c * gp - s * gq;
                        G[i * l + q] = s * gp + c * gq;
                        const float up = U[i * l + p], uq = U[i * l + q];
                        U[i * l + p] = c * up - s * uq;
                        U[i * l + q] = s * up + c * uq;
                    }
                    __syncthreads();
                    for (int i = tid; i < l; i += NTHR) {
                        const float gp = G[p * l + i], gq = G[q * l + i];
                        G[p * l + i] = c * gp - s * gq;
                        G[q * l + i] = s * gp + c * gq;
                    }
                }
                __syncthreads();
            }
        }
    }
}

// ---------------- main kernel: one workgroup per batch sample --------------
__global__ __launch_bounds__(NTHR)
void peps_forward_kernel(const int* __restrict__ configs,
                         const float* __restrict__ tensors,
                         float* __restrict__ out,
                         float* __restrict__ ws)
{
    const int b    = blockIdx.x;
    const int tid  = threadIdx.x;
    const int wv   = tid >> 5;

    __shared__ float red[NWAVE];
    __shared__ float sbc[4];
    __shared__ int   scfg[L1C * L2C];
    __shared__ float eg[160];
    __shared__ int   sidx;
    __shared__ float sT[SITE_SZ];        // site tensor, staged via TDM
    __shared__ float sC[3456];           // cand[j], staged via LDS copy

    float* base = ws + (size_t)b * PER_BATCH;
    float* dblb = base + OFF_DBL;
    float* tmpA = base + OFF_TMP;
    float* Rb   = base + OFF_R;
    float* Gb   = base + OFF_G;
    float* Ub   = base + OFF_U;
    float* Wb   = base + OFF_W;
    float* cb   = base + OFF_CAND;

    for (int i = tid; i < L1C * L2C; i += NTHR) scfg[i] = configs[b * L1C * L2C + i];
    if (tid < 8) __builtin_prefetch(tensors + (size_t)tid * (L1C * L2C * DPH * SITE_SZ / 8), 0, 3);
    __syncthreads();

    int candL[L2C], candR[L2C];

    // ---- row 0 -> initial boundary MPS (l, u=1, m=6, r) ----
    for (int j = 0; j < L2C; ++j) {
        const int lc = (j == 0) ? 1 : DBo;
        const int rc = (j == L2C - 1) ? 1 : DBo;
        const float* T = tensors + ((size_t)(0 * L2C + j) * DPH + scfg[j]) * SITE_SZ;
        float* dst = cb + j * CAND_CAP;
        const int tot = lc * DBo * rc;
        for (int idx = tid; idx < tot; idx += NTHR) {
            const int r_ = idx % rc;
            const int t2 = idx / rc;
            const int m_ = t2 % DBo;
            const int l_ = t2 / DBo;
            dst[idx] = T[l_ * 216 + m_ * 6 + r_];
        }
        candL[j] = lc; candR[j] = rc;
    }
    __syncthreads();

    int la[L2C], lastd[L2C], rj[L2C], Rj[L2C];

    // ---- absorb rows 1..5 ----
    for (int i = 1; i < L1C; ++i) {
        const int rowd = (i == L1C - 1) ? 1 : DBo;

        // (1) double tensors: D[l,L,d,r,R] = sum_m cand[l,m,r] * site[L,m,d,R]
        for (int j = 0; j < L2C; ++j) {
            const int Lr = (j == 0) ? 1 : DBo;
            const int RR = (j == L2C - 1) ? 1 : DBo;
            const int lc = candL[j], rc = candR[j];
            const float* T = tensors + ((size_t)(i * L2C + j) * DPH + scfg[i * L2C + j]) * SITE_SZ;
            const float* cp = cb + j * CAND_CAP;
            // stage: TDM moves the site tensor to LDS while all waves copy cand[j]
            if (wv == 0) tdm_load_lds_1d(sT, T, SITE_SZ);
#if !HAVE_TDM
            for (int idx = tid; idx < SITE_SZ; idx += NTHR) sT[idx] = T[idx];
#endif
            const int csz = lc * DBo * rc;
            for (int idx = tid; idx < csz; idx += NTHR) sC[idx] = cp[idx];
            __syncthreads();

            float* D = dblb + (size_t)j * DBL_CAP;
            const int tot = lc * Lr * rowd * rc * RR;
            for (int idx = tid; idx < tot; idx += NTHR) {
                int t2 = idx;
                const int R_ = t2 % RR;   t2 /= RR;
                const int r_ = t2 % rc;   t2 /= rc;
                const int d_ = t2 % rowd; t2 /= rowd;
                const int Lq = t2 % Lr;
                const int l_ = t2 / Lr;
                float acc = 0.f;
#pragma unroll
                for (int m = 0; m < DBo; ++m)
                    acc += sC[(l_ * DBo + m) * rc + r_] * sT[Lq * 216 + m * 36 + d_ * 6 + R_];
                D[idx] = acc;
            }
            la[j] = lc * Lr; rj[j] = rc; Rj[j] = RR;
            __syncthreads();             // sT/sC reused next j
        }

        // (2) left->right QR sweep
        for (int j = 0; j < L2C - 1; ++j) {
            const int m = la[j] * rowd;
            const int n = rj[j] * Rj[j];
            const int k = (m < n) ? m : n;
            float* A = dblb + (size_t)j * DBL_CAP;
            mgs_qr(A, m, n, k, Rb, red, tid);
            if (k < n) {  // compact Q from stride n to stride k
                for (int idx = tid; idx < m * k; idx += NTHR)
                    tmpA[idx] = A[(idx / k) * n + (idx % k)];
                __syncthreads();
                for (int idx = tid; idx < m * k; idx += NTHR) A[idx] = tmpA[idx];
                __syncthreads();
            }
            lastd[j] = k;
            // fold R into next tensor: (k x n) * (n x rowd*r*R)
            const int c2 = rowd * rj[j + 1] * Rj[j + 1];
            float* B = dblb + (size_t)(j + 1) * DBL_CAP;
            gemm_wmma(Rb, n, 0, B, c2, 0, tmpA, c2, k, c2, n, tid);
            __syncthreads();
            for (int idx = tid; idx < k * c2; idx += NTHR) B[idx] = tmpA[idx];
            __syncthreads();
            la[j + 1] = k;
        }
        lastd[L2C - 1] = rj[L2C - 1] * Rj[L2C - 1];

        // (3) right->left truncated-SVD sweep (Gram + Jacobi; Vh = S^-1 U^T M)
        for (int j = L2C - 1; j >= 1; --j) {
            const int l = la[j];
            const int p = rowd * lastd[j];
            float* Mm = dblb + (size_t)j * DBL_CAP;
            const int ksvd = (l < p) ? l : p;
            const int kp   = (ksvd < DCC) ? ksvd : DCC;

            gemm_wmma(Mm, p, 0, Mm, p, 1, Gb, l, l, l, p, tid);   // G = M M^T
            __syncthreads();
            jacobi_eig(Gb, Ub, l, sbc, tid);
            for (int idx = tid; idx < l; idx += NTHR) eg[idx] = Gb[idx * l + idx];
            __syncthreads();
            // partial selection sort: top-kp eigenpairs to the front
            for (int a = 0; a < kp; ++a) {
                if (tid == 0) {
                    int best = a;
                    for (int q = a + 1; q < l; ++q) if (eg[q] > eg[best]) best = q;
                    sidx = best;
                }
                __syncthreads();
                const int bs = sidx;
                if (bs != a) {
                    if (tid == 0) { float t = eg[a]; eg[a] = eg[bs]; eg[bs] = t; }
                    for (int idx = tid; idx < l; idx += NTHR) {
                        float t = Ub[idx * l + a];
                        Ub[idx * l + a] = Ub[idx * l + bs];
                        Ub[idx * l + bs] = t;
                    }
                }
                __syncthreads();
            }
            // W = U[:, :kp] * S^-1  ->  Vh = W^T M
            for (int idx = tid; idx < l * kp; idx += NTHR) {
                const int c  = idx % kp;
                const float sv = sqrtf(fmaxf(eg[c], 0.f));
                const float iv = (sv > 1e-30f) ? 1.f / sv : 0.f;
                Wb[idx] = Ub[(idx / kp) * l + c] * iv;
            }
            __syncthreads();
            gemm_wmma(Wb, kp, 1, Mm, p, 0, Rb, p, kp, p, l, tid); // Vh (kp x p)
            __syncthreads();
            for (int idx = tid; idx < kp * p; idx += NTHR) Mm[idx] = Rb[idx];
            // W = U[:, :kp] * S  (absorbed into left neighbor)
            for (int idx = tid; idx < l * kp; idx += NTHR) {
                const int c = idx % kp;
                Wb[idx] = Ub[(idx / kp) * l + c] * sqrtf(fmaxf(eg[c], 0.f));
            }
            __syncthreads();
            la[j] = kp;
            const int rows = la[j - 1] * rowd;
            float* P = dblb + (size_t)(j - 1) * DBL_CAP;
            gemm_wmma(P, l, 0, Wb, kp, 0, tmpA, kp, rows, kp, l, tid);
            __syncthreads();
            for (int idx = tid; idx < rows * kp; idx += NTHR) P[idx] = tmpA[idx];
            __syncthreads();
            lastd[j - 1] = kp;
        }

        // (4) result MPS becomes new cand
        for (int j = 0; j < L2C; ++j) {
            const int cnt = la[j] * rowd * lastd[j];
            const float* D = dblb + (size_t)j * DBL_CAP;
            float* dst = cb + j * CAND_CAP;
            for (int idx = tid; idx < cnt; idx += NTHR) dst[idx] = D[idx];
            candL[j] = la[j]; candR[j] = lastd[j];
        }
        __syncthreads();
    }

    // ---- readout: res[b,0,0,0,0] = prod_j cand[j][0,0,0,0] ----
    if (tid == 0) {
        float prod = 1.f;
        for (int j = 0; j < L2C; ++j) prod *= cb[j * CAND_CAP];
        out[b] = prod;
    }
}

extern "C" void kernel_launch(void* const* d_in, const int* in_sizes, int n_in,
                              void* d_out, int out_size, void* d_ws, size_t ws_size,
                              hipStream_t stream)
{
    (void)in_sizes; (void)n_in; (void)out_size; (void)ws_size;
    const int*   configs = (const int*)d_in[0];
    const float* tensors = (const float*)d_in[1];
    float* out = (float*)d_out;
    float* ws  = (float*)d_ws;
    peps_forward_kernel<<<dim3(BATCHC), dim3(NTHR), 0, stream>>>(configs, tensors, out, ws);
}